// LSTMModel_14345190769310
// MI455X (gfx1250) — compile-verified
//
#include <hip/hip_runtime.h>
#include <hip/hip_bf16.h>

// ---------------------------------------------------------------------------
// Model: conv1d(8->16,k3,pad1)+ReLU  ->  LSTM(16->10)  ->  LSTM(10->10)
//        -> sigmoid(linear(h2[:, -1]))
// B=2048, T=512, F=8, C=16, H=10
//
// Kernel 1: fused im2col-conv + input projection (xg1 = relu(conv) @ w_ih1^T
//           + b_ih1 + b_hh1) using V_WMMA_F32_16X16X4_F32.
// Kernel 2: fused two-layer LSTM scan + final linear+sigmoid, lane-per-batch,
//           one wave per WGP, software-pipelined gate loads.
// Workspace: xg1 in d_ws, layout [t][gate][batch] = 512*40*2048 fp32 (160 MiB).
// ---------------------------------------------------------------------------

typedef float v2f __attribute__((ext_vector_type(2)));
typedef float v8f __attribute__((ext_vector_type(8)));

#define B_SZ 2048
#define T_SZ 512
#define F_SZ 8
#define C_SZ 16
#define H_SZ 10
#define G_SZ 40   // 4*H

__device__ __forceinline__ v8f wmma_k4(v8f c, v2f a, v2f b) {
  // D = A(16x4 f32) * B(4x16 f32) + C(16x16 f32)
  return __builtin_amdgcn_wmma_f32_16x16x4_f32(
      false, a, false, b, (short)0, c, false, false);
}

__device__ __forceinline__ float sigmoidf(float v) {
  return 1.0f / (1.0f + __expf(-v));
}

// ---------------------------------------------------------------------------
// Kernel 1: one wave = one tile of 16 batch rows at a fixed timestep t.
// 65536 tiles total (128 batch-tiles x 512 timesteps), 4 waves per block.
// ---------------------------------------------------------------------------
__global__ __launch_bounds__(128) void conv_xg_kernel(
    const float* __restrict__ x,       // (B,T,F)
    const float* __restrict__ conv_w,  // (C,F,3)
    const float* __restrict__ conv_b,  // (C)
    const float* __restrict__ w_ih1,   // (40,16)
    const float* __restrict__ b_ih1,   // (40)
    const float* __restrict__ b_hh1,   // (40)
    float* __restrict__ xg1)           // (T,40,B)
{
  __shared__ float ytile_all[4][16 * 20];  // padded stride 20: conflict-free
  const int wave = threadIdx.x >> 5;
  const int lane = threadIdx.x & 31;
  float* ytile = ytile_all[wave];

  const int tile = blockIdx.x * 4 + wave;  // 0..65535
  const int t    = tile >> 7;              // 0..511
  const int b0   = (tile & 127) << 4;      // batch base
  const int m    = lane & 15;              // row-within-tile / N index
  const int hi   = lane >> 4;              // half-wave select (K split)

  // ---- im2col A tile: A[M=batch][K=f*3+kk] = x[b, t+kk-1, f], zero padded.
  // fp32 WMMA A layout: lanes 0-15 hold K={4s,4s+1}, lanes 16-31 K={4s+2,4s+3}
  const int b = b0 + m;
  v2f a[6];
#pragma unroll
  for (int s = 0; s < 6; ++s) {
#pragma unroll
    for (int j = 0; j < 2; ++j) {
      const int k  = 4 * s + 2 * hi + j;       // im2col column 0..23
      const int f  = k / 3;
      const int kk = k - 3 * f;
      const int tt = t + kk - 1;
      const int tc = tt < 0 ? 0 : (tt > (T_SZ - 1) ? (T_SZ - 1) : tt);
      const float v = x[((b << 9) + tc) * F_SZ + f];   // clamped load
      a[s][j] = ((unsigned)tt < (unsigned)T_SZ) ? v : 0.0f;  // select, no branch
    }
  }

  // ---- conv GEMM: (16x24) x (24x16); B[k][n] = conv_w[n*24 + k]
  v8f acc = {};
#pragma unroll
  for (int s = 0; s < 6; ++s) {
    v2f bm;
#pragma unroll
    for (int j = 0; j < 2; ++j) {
      const int k = 4 * s + 2 * hi + j;
      bm[j] = conv_w[m * 24 + k];
    }
    acc = wmma_k4(acc, a[s], bm);
  }

  // ---- bias + ReLU, write D-layout tile to LDS as [row][chan]
  const float cbias = conv_b[m];  // lane's column = output channel
#pragma unroll
  for (int i = 0; i < 8; ++i) {
    float yv = acc[i] + cbias;
    yv = yv > 0.0f ? yv : 0.0f;
    ytile[(i + 8 * hi) * 20 + m] = yv;  // row = i (lanes 0-15) or i+8 (16-31)
  }
  __syncthreads();  // transpose hand-off (also orders DS ops)

  // ---- reload in A layout: A2[M=row][K=chan]
  v2f a2[4];
#pragma unroll
  for (int s = 0; s < 4; ++s) {
#pragma unroll
    for (int j = 0; j < 2; ++j) {
      const int k = 4 * s + 2 * hi + j;
      a2[s][j] = ytile[m * 20 + k];
    }
  }

  // ---- xg1 = y @ w_ih1^T + (b_ih1 + b_hh1);  N = 40 gates padded to 48
#pragma unroll
  for (int nt = 0; nt < 3; ++nt) {
    const int gN   = nt * 16 + m;
    const int gNc  = gN < (G_SZ - 1) ? gN : (G_SZ - 1);  // clamp, keep EXEC full
    const float w0 = (gN < G_SZ) ? 1.0f : 0.0f;
    v8f acc2 = {};
#pragma unroll
    for (int s = 0; s < 4; ++s) {
      v2f bm;
#pragma unroll
      for (int j = 0; j < 2; ++j) {
        const int k = 4 * s + 2 * hi + j;
        bm[j] = w0 * w_ih1[gNc * C_SZ + k];
      }
      acc2 = wmma_k4(acc2, a2[s], bm);
    }
    if (gN < G_SZ) {
      const float bias = b_ih1[gN] + b_hh1[gN];
#pragma unroll
      for (int i = 0; i < 8; ++i) {
        const int brow = b0 + i + 8 * hi;
        xg1[((long)t * G_SZ + gN) * B_SZ + brow] = acc2[i] + bias;
      }
    }
  }
}

// ---------------------------------------------------------------------------
// Kernel 2: both LSTM layers fused in one streaming scan over t.
// One lane per batch element, one wave (32 lanes) per workgroup so the 64
// waves spread across 64 WGPs (private SIMD issue + L0 + LDS port each).
// State lives in VGPRs; weights broadcast from LDS; gate loads for step t+1
// are issued while step t computes (double-buffered, t-loop unrolled by 2).
// ---------------------------------------------------------------------------
struct LstmState {
  float h1[H_SZ], c1[H_SZ], h2[H_SZ], c2[H_SZ];
};

__device__ __forceinline__ void lstm_two_layer_step(
    const float (&g_in)[G_SZ], LstmState& st,
    const float* __restrict__ s_whh1, const float* __restrict__ s_wih2,
    const float* __restrict__ s_whh2, const float* __restrict__ s_b2) {
  // ----- layer 1: precomputed x-part + recurrent matvec -----
  float g[G_SZ];
#pragma unroll
  for (int j = 0; j < G_SZ; ++j) {
    float s = g_in[j];
#pragma unroll
    for (int k = 0; k < H_SZ; ++k) s = fmaf(s_whh1[j * H_SZ + k], st.h1[k], s);
    g[j] = s;
  }
#pragma unroll
  for (int u = 0; u < H_SZ; ++u) {
    const float ig = sigmoidf(g[u]);
    const float fg = sigmoidf(g[H_SZ + u]);
    const float gg = tanhf(g[2 * H_SZ + u]);
    const float og = sigmoidf(g[3 * H_SZ + u]);
    st.c1[u] = fg * st.c1[u] + ig * gg;
    st.h1[u] = og * tanhf(st.c1[u]);
  }
  // ----- layer 2: both matvecs on the fly -----
  float g2[G_SZ];
#pragma unroll
  for (int j = 0; j < G_SZ; ++j) {
    float s = s_b2[j];
#pragma unroll
    for (int k = 0; k < H_SZ; ++k) {
      s = fmaf(s_wih2[j * H_SZ + k], st.h1[k], s);
      s = fmaf(s_whh2[j * H_SZ + k], st.h2[k], s);
    }
    g2[j] = s;
  }
#pragma unroll
  for (int u = 0; u < H_SZ; ++u) {
    const float ig = sigmoidf(g2[u]);
    const float fg = sigmoidf(g2[H_SZ + u]);
    const float gg = tanhf(g2[2 * H_SZ + u]);
    const float og = sigmoidf(g2[3 * H_SZ + u]);
    st.c2[u] = fg * st.c2[u] + ig * gg;
    st.h2[u] = og * tanhf(st.c2[u]);
  }
}

__global__ __launch_bounds__(32) void lstm_scan_kernel(
    const float* __restrict__ xg1,    // (T,40,B)
    const float* __restrict__ w_hh1,  // (40,10)
    const float* __restrict__ w_ih2,  // (40,10)
    const float* __restrict__ w_hh2,  // (40,10)
    const float* __restrict__ b_ih2,  // (40)
    const float* __restrict__ b_hh2,  // (40)
    const float* __restrict__ lin_w,  // (1,10)
    const float* __restrict__ lin_b,  // (1)
    float* __restrict__ out)          // (B,1)
{
  __shared__ float s_whh1[G_SZ * H_SZ];
  __shared__ float s_wih2[G_SZ * H_SZ];
  __shared__ float s_whh2[G_SZ * H_SZ];
  __shared__ float s_b2[G_SZ];
  __shared__ float s_lw[H_SZ];

  for (int i = threadIdx.x; i < G_SZ * H_SZ; i += 32) {
    s_whh1[i] = w_hh1[i];
    s_wih2[i] = w_ih2[i];
    s_whh2[i] = w_hh2[i];
  }
  for (int i = threadIdx.x; i < G_SZ; i += 32) s_b2[i] = b_ih2[i] + b_hh2[i];
  if (threadIdx.x < H_SZ) s_lw[threadIdx.x] = lin_w[threadIdx.x];
  __syncthreads();  // single-wave workgroup: degenerates to NOP-class sync

  const int b = blockIdx.x * 32 + threadIdx.x;  // 0..2047
  LstmState st;
#pragma unroll
  for (int u = 0; u < H_SZ; ++u) {
    st.h1[u] = 0.f; st.c1[u] = 0.f; st.h2[u] = 0.f; st.c2[u] = 0.f;
  }

  const float* xgb = xg1 + b;

  // -------- software-pipelined scan: double-buffered gate vectors --------
  float ga[G_SZ], gb[G_SZ];
#pragma unroll
  for (int j = 0; j < G_SZ; ++j)                       // prologue: t = 0
    ga[j] = xgb[(long)j * B_SZ];

  for (int t = 0; t < T_SZ; t += 2) {
    // issue loads for step t+1 (always valid: T even), then compute step t
#pragma unroll
    for (int j = 0; j < G_SZ; ++j)
      gb[j] = xgb[((long)(t + 1) * G_SZ + j) * B_SZ];
    lstm_two_layer_step(ga, st, s_whh1, s_wih2, s_whh2, s_b2);

    // issue loads for step t+2 (clamped on last iter; values unused), compute t+1
    const int tn = (t + 2 < T_SZ) ? (t + 2) : 0;       // uniform, branchless
#pragma unroll
    for (int j = 0; j < G_SZ; ++j)
      ga[j] = xgb[((long)tn * G_SZ + j) * B_SZ];
    lstm_two_layer_step(gb, st, s_whh1, s_wih2, s_whh2, s_b2);
  }

  // ----- final linear + sigmoid on h2 at t = T-1 -----
  float v = lin_b[0];
#pragma unroll
  for (int u = 0; u < H_SZ; ++u) v = fmaf(st.h2[u], s_lw[u], v);
  out[b] = sigmoidf(v);
}

// ---------------------------------------------------------------------------
extern "C" void kernel_launch(void* const* d_in, const int* in_sizes, int n_in,
                              void* d_out, int out_size, void* d_ws, size_t ws_size,
                              hipStream_t stream) {
  const float* x      = (const float*)d_in[0];
  const float* conv_w = (const float*)d_in[1];
  const float* conv_b = (const float*)d_in[2];
  const float* w_ih1  = (const float*)d_in[3];
  const float* w_hh1  = (const float*)d_in[4];
  const float* b_ih1  = (const float*)d_in[5];
  const float* b_hh1  = (const float*)d_in[6];
  const float* w_ih2  = (const float*)d_in[7];
  const float* w_hh2  = (const float*)d_in[8];
  const float* b_ih2  = (const float*)d_in[9];
  const float* b_hh2  = (const float*)d_in[10];
  const float* lin_w  = (const float*)d_in[11];
  const float* lin_b  = (const float*)d_in[12];
  float* out = (float*)d_out;
  float* xg1 = (float*)d_ws;  // needs T*40*B*4 = 160 MiB

  // 65536 tiles, 4 waves (tiles) per 128-thread block
  conv_xg_kernel<<<16384, 128, 0, stream>>>(x, conv_w, conv_b,
                                            w_ih1, b_ih1, b_hh1, xg1);
  // 64 single-wave workgroups -> one wave per WGP, 2048 batch lanes
  lstm_scan_kernel<<<64, 32, 0, stream>>>(xg1, w_hh1, w_ih2, w_hh2,
                                          b_ih2, b_hh2, lin_w, lin_b, out);
}